// MultiHeadedSelfAttention_87170656239755
// MI455X (gfx1250) — compile-verified
//
#include <hip/hip_runtime.h>

// ---------------------------------------------------------------------------
// Multi-head self-attention for MI455X (gfx1250), full fp32 via
// V_WMMA_F32_16X16X4_F32.  P=4 B=8 S=512 D=768 H=12 W=64.
// Kernel 1: fused Q/K/V projection GEMM (x @ W^T + b), V stored transposed.
// Kernel 2: flash-attention (online softmax) per 16 query rows per wave.
// ---------------------------------------------------------------------------

typedef float v2f __attribute__((ext_vector_type(2)));
typedef float v8f __attribute__((ext_vector_type(8)));

#define Pn 4
#define Bn 8
#define Sn 512
#define Dn 768
#define Hn 12
#define Wn 64
#define MROWS (Pn * Bn * Sn) /* 16384 */

static __device__ __forceinline__ v8f wmma_f32(v2f a, v2f b, v8f c) {
  // D = A(16x4 f32) * B(4x16 f32) + C(16x16 f32)
  return __builtin_amdgcn_wmma_f32_16x16x4_f32(
      /*neg_a=*/false, a, /*neg_b=*/false, b,
      /*c_mod=*/(short)0, c, /*reuse_a=*/false, /*reuse_b=*/false);
}

// Store one 16x16 f32 C-tile into [pb, h, S, 64] layout.
static __device__ __forceinline__ void store_qk(float* base, int pb, int sBase,
                                                int nbase, int l15, int lh,
                                                const v8f& acc) {
  const int h = nbase >> 6;
  const int w = (nbase & 63) + l15;
  float* p = base + ((size_t)pb * Hn + h) * (size_t)Sn * Wn;
#pragma unroll
  for (int r = 0; r < 8; ++r) {
    const int s = sBase + r + 8 * lh;
    p[(size_t)s * Wn + w] = acc[r];
  }
}

// Store one 16x16 f32 C-tile transposed into [pb, h, 64, S] layout (for V).
static __device__ __forceinline__ void store_vT(float* base, int pb, int sBase,
                                                int nbase, int l15, int lh,
                                                const v8f& acc) {
  const int h = nbase >> 6;
  const int w = (nbase & 63) + l15;
  float* p = base + (((size_t)pb * Hn + h) * Wn + w) * (size_t)Sn;
#pragma unroll
  for (int r = 0; r < 8; ++r) {
    const int s = sBase + r + 8 * lh;
    p[s] = acc[r];
  }
}

// ---------------------------------------------------------------------------
// Kernel 1: Q/K/V = x @ W{q,k,v}^T + b{q,k,v}
// One wave computes a 16(M) x 32(N) tile for Q, K and V (6 accumulators),
// sharing the A (x) fragment across all six WMMAs each K-step.
// ---------------------------------------------------------------------------
__global__ __launch_bounds__(128) void qkv_proj(
    const float* __restrict__ x, const float* __restrict__ Wq,
    const float* __restrict__ bq, const float* __restrict__ Wk,
    const float* __restrict__ bk, const float* __restrict__ Wv,
    const float* __restrict__ bv, float* __restrict__ qws,
    float* __restrict__ kws, float* __restrict__ vws) {
  const int lane = threadIdx.x & 31;
  const int wid = blockIdx.x * (blockDim.x >> 5) + (threadIdx.x >> 5);
  const int mTile = wid & (MROWS / 16 - 1);  // 0..1023
  const int nGrp = wid >> 10;                // 0..23
  const int m0 = mTile * 16;
  const int n0 = nGrp * 32;
  const int l15 = lane & 15;
  const int lh = lane >> 4;

  // fp32 WMMA fragment addressing: lane (l15,lh) holds a contiguous float2
  // at row l15, cols {k0+2*lh, k0+2*lh+1} for both A(row-major x) and
  // B(row-major weight, since q = x @ W^T).
  const float* ap = x + (size_t)(m0 + l15) * Dn + 2 * lh;
  const float* pq0 = Wq + (size_t)(n0 + l15) * Dn + 2 * lh;
  const float* pq1 = Wq + (size_t)(n0 + 16 + l15) * Dn + 2 * lh;
  const float* pk0 = Wk + (size_t)(n0 + l15) * Dn + 2 * lh;
  const float* pk1 = Wk + (size_t)(n0 + 16 + l15) * Dn + 2 * lh;
  const float* pv0 = Wv + (size_t)(n0 + l15) * Dn + 2 * lh;
  const float* pv1 = Wv + (size_t)(n0 + 16 + l15) * Dn + 2 * lh;

  v8f aq0 = {}, aq1 = {}, ak0 = {}, ak1 = {}, av0 = {}, av1 = {};
  for (int k0 = 0; k0 < Dn; k0 += 4) {
    const v2f a = *(const v2f*)(ap + k0);
    aq0 = wmma_f32(a, *(const v2f*)(pq0 + k0), aq0);
    aq1 = wmma_f32(a, *(const v2f*)(pq1 + k0), aq1);
    ak0 = wmma_f32(a, *(const v2f*)(pk0 + k0), ak0);
    ak1 = wmma_f32(a, *(const v2f*)(pk1 + k0), ak1);
    av0 = wmma_f32(a, *(const v2f*)(pv0 + k0), av0);
    av1 = wmma_f32(a, *(const v2f*)(pv1 + k0), av1);
  }

  // Bias (column-constant per lane).
  const float biq0 = bq[n0 + l15], biq1 = bq[n0 + 16 + l15];
  const float bik0 = bk[n0 + l15], bik1 = bk[n0 + 16 + l15];
  const float biv0 = bv[n0 + l15], biv1 = bv[n0 + 16 + l15];
#pragma unroll
  for (int r = 0; r < 8; ++r) {
    aq0[r] += biq0; aq1[r] += biq1;
    ak0[r] += bik0; ak1[r] += bik1;
    av0[r] += biv0; av1[r] += biv1;
  }

  const int pb = m0 >> 9;         // 512 rows per (p,b)
  const int sBase = m0 & (Sn - 1);
  store_qk(qws, pb, sBase, n0, l15, lh, aq0);
  store_qk(qws, pb, sBase, n0 + 16, l15, lh, aq1);
  store_qk(kws, pb, sBase, n0, l15, lh, ak0);
  store_qk(kws, pb, sBase, n0 + 16, l15, lh, ak1);
  store_vT(vws, pb, sBase, n0, l15, lh, av0);
  store_vT(vws, pb, sBase, n0 + 16, l15, lh, av1);
}

// ---------------------------------------------------------------------------
// Kernel 2: flash attention.  One wave = 16 query rows of one head.
// scores = (q*0.125) @ k^T  + mask penalty ; online softmax ; O += P @ V.
// P (C-layout) -> A-layout via a tiny padded LDS tile + s_wait_dscnt.
// ---------------------------------------------------------------------------
__global__ __launch_bounds__(128) void attn(const float* __restrict__ qws,
                                            const float* __restrict__ kws,
                                            const float* __restrict__ vws,
                                            const int* __restrict__ mask,
                                            float* __restrict__ out) {
  __shared__ float lds[4][16][18];  // 18-float row pitch: 8B aligned, no bank conflicts
  const int lane = threadIdx.x & 31;
  const int warp = threadIdx.x >> 5;
  const int wid = blockIdx.x * 4 + warp;
  const int head = wid >> 5;   // 0..383  == pb*12 + h
  const int sT = wid & 31;     // 0..31
  const int pb = head / Hn;
  const int h = head % Hn;
  const int b = pb & (Bn - 1);
  const int s0 = sT * 16;
  const int l15 = lane & 15;
  const int lh = lane >> 4;

  const float* qbase = qws + (size_t)head * Sn * Wn;
  const float* kbase = kws + (size_t)head * Sn * Wn;
  const float* vbase = vws + (size_t)head * Wn * Sn;  // transposed [64][512]
  const int* mrow = mask + b * Sn;

  // Resident A-fragments of q (scaled by 1/sqrt(64)).
  v2f aq[16];
  {
    const float* qp = qbase + (size_t)(s0 + l15) * Wn + 2 * lh;
#pragma unroll
    for (int kk = 0; kk < 16; ++kk) {
      v2f t = *(const v2f*)(qp + kk * 4);
      aq[kk] = t * 0.125f;
    }
  }

  v8f O0 = {}, O1 = {}, O2 = {}, O3 = {};
  float mr[8], lr[8];
#pragma unroll
  for (int r = 0; r < 8; ++r) {
    mr[r] = -__builtin_inff();
    lr[r] = 0.0f;
  }

  float(*pl)[18] = lds[warp];

  for (int t0 = 0; t0 < Sn; t0 += 16) {
    // Mask penalty for this lane's key column.
    const float pen = -10000.0f * (1.0f - (float)mrow[t0 + l15]);

    // scores tile: 16 WMMAs over K=W=64.
    v8f C = {};
    const float* kp = kbase + (size_t)(t0 + l15) * Wn + 2 * lh;
#pragma unroll
    for (int kk = 0; kk < 16; ++kk)
      C = wmma_f32(aq[kk], *(const v2f*)(kp + kk * 4), C);

    // Online softmax: row reductions live within each 16-lane half.
    float sc[8];
#pragma unroll
    for (int r = 0; r < 8; ++r) {
      float c = C[r] + pen;
      float t = c;
      t = fmaxf(t, __shfl_xor(t, 1, 32));
      t = fmaxf(t, __shfl_xor(t, 2, 32));
      t = fmaxf(t, __shfl_xor(t, 4, 32));
      t = fmaxf(t, __shfl_xor(t, 8, 32));
      const float nm = fmaxf(mr[r], t);
      sc[r] = __expf(mr[r] - nm);
      mr[r] = nm;
      const float p = __expf(c - nm);
      C[r] = p;
      float s = p;
      s += __shfl_xor(s, 1, 32);
      s += __shfl_xor(s, 2, 32);
      s += __shfl_xor(s, 4, 32);
      s += __shfl_xor(s, 8, 32);
      lr[r] = lr[r] * sc[r] + s;
    }

    // C-layout -> A-layout for P via LDS (wave-private tile).
#pragma unroll
    for (int r = 0; r < 8; ++r) pl[r + 8 * lh][l15] = C[r];
    asm volatile("s_wait_dscnt 0x0" ::: "memory");
    v2f pf[4];
#pragma unroll
    for (int kk = 0; kk < 4; ++kk)
      pf[kk] = *(const v2f*)&pl[l15][kk * 4 + 2 * lh];
    asm volatile("s_wait_dscnt 0x0" ::: "memory");

    // Rescale O by row, then O += P @ V  (V read from transposed layout).
#pragma unroll
    for (int r = 0; r < 8; ++r) {
      O0[r] *= sc[r];
      O1[r] *= sc[r];
      O2[r] *= sc[r];
      O3[r] *= sc[r];
    }
    const float* vp = vbase + (size_t)l15 * Sn + t0 + 2 * lh;
#pragma unroll
    for (int kk = 0; kk < 4; ++kk) {
      O0 = wmma_f32(pf[kk], *(const v2f*)(vp + kk * 4), O0);
      O1 = wmma_f32(pf[kk], *(const v2f*)(vp + 16 * Sn + kk * 4), O1);
      O2 = wmma_f32(pf[kk], *(const v2f*)(vp + 32 * Sn + kk * 4), O2);
      O3 = wmma_f32(pf[kk], *(const v2f*)(vp + 48 * Sn + kk * 4), O3);
    }
  }

  // Normalize and write back to [P,B,S,D].
#pragma unroll
  for (int r = 0; r < 8; ++r) {
    const float inv = 1.0f / lr[r];
    const size_t row =
        ((size_t)pb * Sn + (size_t)(s0 + r + 8 * lh)) * Dn + (size_t)h * Wn;
    out[row + 0 + l15] = O0[r] * inv;
    out[row + 16 + l15] = O1[r] * inv;
    out[row + 32 + l15] = O2[r] * inv;
    out[row + 48 + l15] = O3[r] * inv;
  }
}

// ---------------------------------------------------------------------------
extern "C" void kernel_launch(void* const* d_in, const int* in_sizes, int n_in,
                              void* d_out, int out_size, void* d_ws,
                              size_t ws_size, hipStream_t stream) {
  const float* x = (const float*)d_in[0];
  const float* Wq = (const float*)d_in[1];
  const float* bq = (const float*)d_in[2];
  const float* Wk = (const float*)d_in[3];
  const float* bk = (const float*)d_in[4];
  const float* Wv = (const float*)d_in[5];
  const float* bv = (const float*)d_in[6];
  const int* mask = (const int*)d_in[7];
  float* out = (float*)d_out;

  const size_t per = (size_t)Pn * Bn * Hn * Sn * Wn;  // 12,582,912 floats
  float* qws = (float*)d_ws;
  float* kws = qws + per;
  float* vws = kws + per;

  // Kernel 1: 1024 M-tiles x 24 N-groups = 24576 waves, 4 waves/block.
  const int waves1 = (MROWS / 16) * (Dn / 32);
  qkv_proj<<<waves1 / 4, dim3(128), 0, stream>>>(x, Wq, bq, Wk, bk, Wv, bv,
                                                 qws, kws, vws);

  // Kernel 2: 384 heads x 32 s-tiles = 12288 waves, 4 waves/block.
  const int waves2 = (Pn * Bn * Hn) * (Sn / 16);
  attn<<<waves2 / 4, dim3(128), 0, stream>>>(qws, kws, vws, mask, out);
}